// irnn_70119636075160
// MI455X (gfx1250) — compile-verified
//
#include <hip/hip_runtime.h>
#include <cstdint>
#include <cstddef>

// Problem shape (fixed by the reference harness).
static constexpr int B_  = 4;
static constexpr int C_  = 64;
static constexpr int H_  = 256;
static constexpr int W_  = 256;
static constexpr int TPB = 256;   // threads per block
static constexpr int CW  = 64;    // chunk width (columns staged per LDS tile)
static constexpr int LSTR = 66;   // LDS row stride in floats: (2*lane+col)%64 -> conflict-free b64

typedef float v4f __attribute__((ext_vector_type(4)));

// ---------------- CDNA5 async global<->LDS helpers (ASYNCcnt path) ----------------

__device__ __forceinline__ uint32_t lds_off(const void* p) {
  // Low 32 bits of a flat shared-aperture address are the LDS byte offset
  // (ISA 10.2).
  return (uint32_t)(uintptr_t)p;
}

__device__ __forceinline__ void async_ld_b64(uint32_t lds, const float* g) {
  asm volatile("global_load_async_to_lds_b64 %0, %1, off"
               :: "v"(lds), "v"((uint64_t)(uintptr_t)g) : "memory");
}

// Outputs are write-once / never re-read: non-temporal store keeps L2 for x.
__device__ __forceinline__ void async_st_b64_nt(float* g, uint32_t lds) {
  asm volatile("global_store_async_from_lds_b64 %0, %1, off th:TH_STORE_NT"
               :: "v"((uint64_t)(uintptr_t)g), "v"(lds) : "memory");
}

__device__ __forceinline__ void wait_async0() {
  asm volatile("s_wait_asynccnt 0" ::: "memory");
}

// ---------------- small math helpers ----------------

__device__ __forceinline__ v4f relu4(v4f v) {
  v.x = fmaxf(v.x, 0.0f); v.y = fmaxf(v.y, 0.0f);
  v.z = fmaxf(v.z, 0.0f); v.w = fmaxf(v.w, 0.0f);
  return v;
}

__device__ __forceinline__ v4f step4(float w, float b, v4f h, v4f xv) {
  h.x = fmaxf(w * h.x + (b + xv.x), 0.0f);
  h.y = fmaxf(w * h.y + (b + xv.y), 0.0f);
  h.z = fmaxf(w * h.z + (b + xv.z), 0.0f);
  h.w = fmaxf(w * h.w + (b + xv.w), 0.0f);
  return h;
}

// ---------------- vertical scans (down & up): naturally coalesced ----------------
// One thread owns 4 consecutive columns (v4f) of one (b,c) plane.
// Direction split across blocks: blocks [0,64) = down, [64,128) = up.
// x fits in the 192 MB L2, so both directions' reads are L2-resident.

__global__ __launch_bounds__(TPB) void irnn_v(
    const float* __restrict__ x,
    const float* __restrict__ wd_, const float* __restrict__ bd_,
    const float* __restrict__ wu_, const float* __restrict__ bu_,
    float* __restrict__ o_down, float* __restrict__ o_up)
{
  const int nseg_blocks = (B_ * C_ * W_ / 4) / TPB;          // 64
  const bool is_up = blockIdx.x >= (unsigned)nseg_blocks;
  const int  blk   = blockIdx.x - (is_up ? nseg_blocks : 0);

  const int gid   = blk * TPB + threadIdx.x;                 // 0 .. B*C*W/4 - 1
  const int plane = gid >> 6;                                // b*C + c  (W/4 = 64 segs/plane)
  const int seg   = gid & 63;
  const int ch    = plane & (C_ - 1);
  const size_t base = (size_t)plane * (H_ * W_) + (size_t)seg * 4;

  if (!is_up) {
    const float w = wd_[ch], b = bd_[ch];
    v4f hv = relu4(*(const v4f*)(x + base));
    __builtin_nontemporal_store(hv, (v4f*)(o_down + base));
    for (int r = 1; r < H_; ++r) {
      const size_t off = base + (size_t)r * W_;
      hv = step4(w, b, hv, *(const v4f*)(x + off));
      __builtin_nontemporal_store(hv, (v4f*)(o_down + off));
    }
  } else {
    const float w = wu_[ch], b = bu_[ch];
    const size_t top = base + (size_t)(H_ - 1) * W_;
    v4f hv = relu4(*(const v4f*)(x + top));
    __builtin_nontemporal_store(hv, (v4f*)(o_up + top));
    for (int r = H_ - 2; r >= 0; --r) {
      const size_t off = base + (size_t)r * W_;
      hv = step4(w, b, hv, *(const v4f*)(x + off));
      __builtin_nontemporal_store(hv, (v4f*)(o_up + off));
    }
  }
}

// ---------------- horizontal scans (right & left): LDS-staged via async DMA ----------------
// Block = 256 threads = 256 rows (rows flattened over B*C*H). W is processed in
// 4 chunks of 64 columns; each chunk is DMA'd global->LDS coalesced, scanned
// per-row out of LDS (stride 66 -> bank (2*lane+col)%64, conflict-free b64),
// results staged in an LDS out tile and DMA'd LDS->global coalesced (NT).
// Direction split across blocks: [0,256) = right, [256,512) = left.

__global__ __launch_bounds__(TPB) void irnn_h(
    const float* __restrict__ x,
    const float* __restrict__ wr_, const float* __restrict__ br_,
    const float* __restrict__ wl_, const float* __restrict__ bl_,
    float* __restrict__ o_right, float* __restrict__ o_left)
{
  __shared__ __align__(16) float xin[TPB * LSTR];   // 66 KiB input tile
  __shared__ __align__(16) float oti[TPB * LSTR];   // 66 KiB output tile

  const int nrow_blocks = (B_ * C_ * H_) / TPB;     // 256
  const bool is_left = blockIdx.x >= (unsigned)nrow_blocks;
  const int  blk     = blockIdx.x - (is_left ? nrow_blocks : 0);

  const int t   = threadIdx.x;
  const int r0  = blk * TPB;                        // first flattened row of block
  const int row = r0 + t;
  const int ch  = (row / H_) & (C_ - 1);            // row = (b*C + c)*H + h

  const uint32_t xb = lds_off(xin);
  const uint32_t ob = lds_off(oti);

  // Coalesced staging of a 256-row x 64-col chunk (float2 granularity, 32 ops/thread).
  auto stage_in = [&](int c0) {
    #pragma unroll
    for (int i = 0; i < 32; ++i) {
      const int q  = i * TPB + t;       // float2 index in chunk (32 f2 per row)
      const int rr = q >> 5;
      const int cc = (q & 31) << 1;
      async_ld_b64(xb + (uint32_t)(rr * LSTR + cc) * 4u,
                   x + (size_t)(r0 + rr) * W_ + c0 + cc);
    }
  };
  auto stage_out = [&](float* dst, int c0) {
    #pragma unroll
    for (int i = 0; i < 32; ++i) {
      const int q  = i * TPB + t;
      const int rr = q >> 5;
      const int cc = (q & 31) << 1;
      async_st_b64_nt(dst + (size_t)(r0 + rr) * W_ + c0 + cc,
                      ob + (uint32_t)(rr * LSTR + cc) * 4u);
    }
  };

  float h = 0.0f;

  if (!is_left) {
    const float w = wr_[ch], b = br_[ch];
    // RIGHT: forward over W. h_0 = relu(x_0).
    for (int k = 0; k < W_ / CW; ++k) {
      const int c0 = k * CW;
      stage_in(c0);
      wait_async0();        // waits own loads AND previous chunk's out-tile stores
      __syncthreads();      // barrier => all waves' async ops drained before compute
      #pragma unroll
      for (int cc = 0; cc < CW; cc += 2) {
        const float2 xv = *(const float2*)&xin[t * LSTR + cc];
        const float h0 = (cc == 0 && k == 0) ? fmaxf(xv.x, 0.0f)
                                             : fmaxf(w * h  + (b + xv.x), 0.0f);
        const float h1 = fmaxf(w * h0 + (b + xv.y), 0.0f);
        *(float2*)&oti[t * LSTR + cc] = make_float2(h0, h1);
        h = h1;
      }
      __syncthreads();      // out tile complete & visible
      stage_out(o_right, c0);
    }
  } else {
    const float w = wl_[ch], b = bl_[ch];
    // LEFT: backward over W. h_{W-1} = relu(x_{W-1}).
    for (int k = W_ / CW - 1; k >= 0; --k) {
      const int c0 = k * CW;
      stage_in(c0);
      wait_async0();
      __syncthreads();
      #pragma unroll
      for (int cc = CW - 2; cc >= 0; cc -= 2) {
        const float2 xv = *(const float2*)&xin[t * LSTR + cc];
        const float h1 = (cc == CW - 2 && k == W_ / CW - 1)
                           ? fmaxf(xv.y, 0.0f)
                           : fmaxf(w * h  + (b + xv.y), 0.0f);
        const float h0 = fmaxf(w * h1 + (b + xv.x), 0.0f);
        *(float2*)&oti[t * LSTR + cc] = make_float2(h0, h1);
        h = h0;
      }
      __syncthreads();
      stage_out(o_left, c0);
    }
  }

  wait_async0();            // drain final stores (S_ENDPGM also implies wait-idle)
}

// ---------------- launcher ----------------
// d_in order: input_feature, w_up, b_up, w_right, b_right, w_down, b_down, w_left, b_left
// d_out order: out_up | out_right | out_down | out_left, each B*C*H*W floats.

extern "C" void kernel_launch(void* const* d_in, const int* in_sizes, int n_in,
                              void* d_out, int out_size, void* d_ws, size_t ws_size,
                              hipStream_t stream) {
  (void)in_sizes; (void)n_in; (void)out_size; (void)d_ws; (void)ws_size;

  const float* x       = (const float*)d_in[0];
  const float* w_up    = (const float*)d_in[1];
  const float* b_up    = (const float*)d_in[2];
  const float* w_right = (const float*)d_in[3];
  const float* b_right = (const float*)d_in[4];
  const float* w_down  = (const float*)d_in[5];
  const float* b_down  = (const float*)d_in[6];
  const float* w_left  = (const float*)d_in[7];
  const float* b_left  = (const float*)d_in[8];

  float* out = (float*)d_out;
  const size_t N = (size_t)B_ * C_ * H_ * W_;

  float* out_up    = out + 0 * N;
  float* out_right = out + 1 * N;
  float* out_down  = out + 2 * N;
  float* out_left  = out + 3 * N;

  // vertical: down blocks [0,64) + up blocks [64,128)
  irnn_v<<<2 * ((B_ * C_ * W_ / 4) / TPB), TPB, 0, stream>>>(
      x, w_down, b_down, w_up, b_up, out_down, out_up);

  // horizontal: right blocks [0,256) + left blocks [256,512)
  irnn_h<<<2 * ((B_ * C_ * H_) / TPB), TPB, 0, stream>>>(
      x, w_right, b_right, w_left, b_left, out_right, out_left);
}